// Controller_13597866459792
// MI455X (gfx1250) — compile-verified
//
#include <hip/hip_runtime.h>
#include <hip/hip_bf16.h>

typedef float v2f __attribute__((ext_vector_type(2)));
typedef float v8f __attribute__((ext_vector_type(8)));
typedef int   i32x4 __attribute__((vector_size(16)));

#define TILE_M 128
#define TILE_N 128
#define TILE_K 32

#if __has_builtin(__builtin_amdgcn_global_load_async_to_lds_b128) && \
    __has_builtin(__builtin_amdgcn_global_load_async_to_lds_b32)  && \
    __has_builtin(__builtin_amdgcn_s_wait_asynccnt)
#define USE_ASYNC 1
#else
#define USE_ASYNC 0
#endif

// ---- async global->LDS copies (CDNA5 ASYNCcnt path), sync fallback ----
__device__ __forceinline__ void cp_b128(const float* g, float* l) {
#if USE_ASYNC
  __builtin_amdgcn_global_load_async_to_lds_b128(
      (__attribute__((address_space(1))) i32x4*)g,
      (__attribute__((address_space(3))) i32x4*)l, 0, 0);
#else
  *(float4*)l = *(const float4*)g;
#endif
}
__device__ __forceinline__ void cp_b32(const float* g, float* l) {
#if USE_ASYNC
  __builtin_amdgcn_global_load_async_to_lds_b32(
      (__attribute__((address_space(1))) int*)g,
      (__attribute__((address_space(3))) int*)l, 0, 0);
#else
  *l = *g;
#endif
}
template <int Nw> __device__ __forceinline__ void wait_async() {
#if USE_ASYNC
  __builtin_amdgcn_s_wait_asynccnt(Nw);
#endif
}

// ---- full-tile staging: As[k][m] = opA(m_blk+m, k0+k) ----
template <int TA>
__device__ __forceinline__ void stage_A(const float* __restrict__ A, float* dst,
                                        int m_blk, int k0, int lda, int tid)
{
  if (TA) {  // global fast index = m, LDS fast index = m -> b128
    #pragma unroll
    for (int it = 0; it < 4; ++it) {
      int idx = tid + it * 256;
      int m4 = idx & 31, kk = idx >> 5;
      cp_b128(A + (size_t)(k0 + kk) * lda + m_blk + m4 * 4,
              dst + kk * TILE_M + m4 * 4);
    }
  } else {   // global fast = k, LDS fast = m -> per-element b32 (DMA scatter)
    #pragma unroll
    for (int it = 0; it < 16; ++it) {
      int idx = tid + it * 256;
      int kk = idx & (TILE_K - 1), m = idx >> 5;
      cp_b32(A + (size_t)(m_blk + m) * lda + k0 + kk,
             dst + kk * TILE_M + m);
    }
  }
}
template <int TB>
__device__ __forceinline__ void stage_B(const float* __restrict__ B, float* dst,
                                        int n_blk, int k0, int ldb, int tid)
{
  if (!TB) { // global fast = n, LDS fast = n -> b128
    #pragma unroll
    for (int it = 0; it < 4; ++it) {
      int idx = tid + it * 256;
      int n4 = idx & 31, kk = idx >> 5;
      cp_b128(B + (size_t)(k0 + kk) * ldb + n_blk + n4 * 4,
              dst + kk * TILE_N + n4 * 4);
    }
  } else {   // global fast = k -> per-element b32
    #pragma unroll
    for (int it = 0; it < 16; ++it) {
      int idx = tid + it * 256;
      int kk = idx & (TILE_K - 1), n = idx >> 5;
      cp_b32(B + (size_t)(n_blk + n) * ldb + k0 + kk,
             dst + kk * TILE_N + n);
    }
  }
}

// ---- one K-tile of WMMA work: 8 k-steps x 8 frags = 64 v_wmma ----
__device__ __forceinline__ void wmma_tile(const float* Asb, const float* Bsb,
                                          v8f (&acc)[2][4], int wm0, int wn0,
                                          int half, int lr)
{
  #pragma unroll
  for (int kk = 0; kk < TILE_K; kk += 4) {
    v2f af[2], bf[4];
    #pragma unroll
    for (int i = 0; i < 2; ++i) {
      int m = wm0 + i * 16 + lr;
      af[i].x = Asb[(kk + 2 * half + 0) * TILE_M + m];
      af[i].y = Asb[(kk + 2 * half + 1) * TILE_M + m];
    }
    #pragma unroll
    for (int j = 0; j < 4; ++j) {
      int n = wn0 + j * 16 + lr;
      bf[j].x = Bsb[(kk + 2 * half + 0) * TILE_N + n];
      bf[j].y = Bsb[(kk + 2 * half + 1) * TILE_N + n];
    }
    #pragma unroll
    for (int i = 0; i < 2; ++i)
      #pragma unroll
      for (int j = 0; j < 4; ++j)
        acc[i][j] = __builtin_amdgcn_wmma_f32_16x16x4_f32(
            false, af[i], false, bf[j], (short)0, acc[i][j], false, false);
  }
}

// ---------------------------------------------------------------------------
// Generic f32 WMMA GEMM:  C = alpha * opA(A) @ opB(B) + beta * C
//   opA(A)[m,k] = TA ? A[k*lda + m] : A[m*lda + k]
//   opB(B)[k,n] = TB ? B[n*ldb + k] : B[k*ldb + n]
// 256 threads = 8 waves; WG tile 128x128; wave tile 32x64 (2x4 WMMA frags).
// Full blocks: double-buffered GLOBAL_LOAD_ASYNC_TO_LDS with counted
// s_wait_asynccnt; edge blocks: guarded synchronous staging.
// ---------------------------------------------------------------------------
template <int TA, int TB>
__global__ __launch_bounds__(256) void gemm_wmma_f32(
    const float* __restrict__ A, const float* __restrict__ B,
    float* __restrict__ C,
    int M, int N, int K, int lda, int ldb, int ldc,
    float alpha, float beta)
{
  __shared__ alignas(16) float As[2][TILE_K * TILE_M];
  __shared__ alignas(16) float Bs[2][TILE_K * TILE_N];
  const int tid  = threadIdx.x;
  const int wave = tid >> 5;
  const int lane = tid & 31;
  const int half = lane >> 4;
  const int lr   = lane & 15;
  const int m_blk = blockIdx.y * TILE_M;
  const int n_blk = blockIdx.x * TILE_N;
  const int wm0 = (wave >> 1) * 32;
  const int wn0 = (wave & 1)  * 64;

  v8f acc[2][4];
  #pragma unroll
  for (int i = 0; i < 2; ++i)
    #pragma unroll
    for (int j = 0; j < 4; ++j)
      #pragma unroll
      for (int v = 0; v < 8; ++v) acc[i][j][v] = 0.0f;

  const bool full = (m_blk + TILE_M <= M) && (n_blk + TILE_N <= N) &&
                    (K % TILE_K == 0);

  if (full) {
    constexpr int NL = (TA ? 4 : 16) + (TB ? 16 : 4);  // async issues per thread/tile
    stage_A<TA>(A, As[0], m_blk, 0, lda, tid);
    stage_B<TB>(B, Bs[0], n_blk, 0, ldb, tid);
    const int nt = K / TILE_K;
    for (int kt = 0; kt < nt; ++kt) {
      const int cur = kt & 1;
      if (kt > 0) __syncthreads();            // prev compute done -> next buf free
      if (kt + 1 < nt) {
        stage_A<TA>(A, As[1 - cur], m_blk, (kt + 1) * TILE_K, lda, tid);
        stage_B<TB>(B, Bs[1 - cur], n_blk, (kt + 1) * TILE_K, ldb, tid);
        wait_async<NL>();                     // drain current tile, leave next in flight
      } else {
        wait_async<0>();
      }
      __syncthreads();                        // current tile visible to all waves
      wmma_tile(As[cur], Bs[cur], acc, wm0, wn0, half, lr);
    }
  } else {
    for (int k0 = 0; k0 < K; k0 += TILE_K) {
      if (TA) {
        #pragma unroll
        for (int it = 0; it < (TILE_K * TILE_M) / 256; ++it) {
          int idx = tid + it * 256;
          int m = idx & (TILE_M - 1), kk = idx >> 7;
          int gm = m_blk + m, gk = k0 + kk;
          float v = 0.0f;
          if (gm < M && gk < K) v = A[(size_t)gk * lda + gm];
          As[0][kk * TILE_M + m] = v;
        }
      } else {
        #pragma unroll
        for (int it = 0; it < (TILE_K * TILE_M) / 256; ++it) {
          int idx = tid + it * 256;
          int kk = idx & (TILE_K - 1), m = idx >> 5;
          int gm = m_blk + m, gk = k0 + kk;
          float v = 0.0f;
          if (gm < M && gk < K) v = A[(size_t)gm * lda + gk];
          As[0][kk * TILE_M + m] = v;
        }
      }
      if (TB) {
        #pragma unroll
        for (int it = 0; it < (TILE_K * TILE_N) / 256; ++it) {
          int idx = tid + it * 256;
          int kk = idx & (TILE_K - 1), n = idx >> 5;
          int gn = n_blk + n, gk = k0 + kk;
          float v = 0.0f;
          if (gn < N && gk < K) v = B[(size_t)gn * ldb + gk];
          Bs[0][kk * TILE_N + n] = v;
        }
      } else {
        #pragma unroll
        for (int it = 0; it < (TILE_K * TILE_N) / 256; ++it) {
          int idx = tid + it * 256;
          int n = idx & (TILE_N - 1), kk = idx >> 7;
          int gn = n_blk + n, gk = k0 + kk;
          float v = 0.0f;
          if (gn < N && gk < K) v = B[(size_t)gk * ldb + gn];
          Bs[0][kk * TILE_N + n] = v;
        }
      }
      __syncthreads();
      wmma_tile(As[0], Bs[0], acc, wm0, wn0, half, lr);
      __syncthreads();
    }
  }

  // epilogue: C/D layout — VGPR v: rows v (lanes 0-15) / v+8 (lanes 16-31)
  #pragma unroll
  for (int i = 0; i < 2; ++i)
    #pragma unroll
    for (int j = 0; j < 4; ++j)
      #pragma unroll
      for (int v = 0; v < 8; ++v) {
        int gm = m_blk + wm0 + i * 16 + 8 * half + v;
        int gn = n_blk + wn0 + j * 16 + lr;
        if (gm < M && gn < N) {
          float r = alpha * acc[i][j][v];
          if (beta != 0.0f) r += beta * C[(size_t)gm * ldc + gn];
          C[(size_t)gm * ldc + gn] = r;
        }
      }
}

// ---------------------------------------------------------------------------
__global__ void build_E_kernel(float* __restrict__ E, const float* __restrict__ G11,
                               const float* __restrict__ G33, const float* __restrict__ Y,
                               int n)
{
  int idx = blockIdx.x * 256 + threadIdx.x;
  if (idx >= n * n) return;
  int i = idx / n, j = idx - i * n;
  float e = 0.5f * (G11[idx] + G33[idx] + Y[idx] - Y[(size_t)j * n + i]);
  if (i == j) e += 1e-3f;
  E[idx] = e;
}

__global__ void add_inplace_kernel(float* __restrict__ dst, const float* __restrict__ src, int cnt)
{
  int idx = blockIdx.x * 256 + threadIdx.x;
  if (idx < cnt) dst[idx] += src[idx];
}

__global__ void copy_kernel(float* __restrict__ dst, const float* __restrict__ src, int cnt)
{
  int idx = blockIdx.x * 256 + threadIdx.x;
  if (idx < cnt) dst[idx] = src[idx];
}

// ---------------------------------------------------------------------------
// Sequential tanh recurrence (lax.scan), independent per batch row:
//   eps[b,i] = tanh((P[b,i] - sum_{j<i} G22[i,j]*eps[b,j]) / (0.5*(G22[i,i]+eps)))
// One wave per batch row (8 rows/WG); eps lives in LDS; G22 row staged per step.
// ---------------------------------------------------------------------------
__global__ __launch_bounds__(256) void scan_eps_kernel(
    const float* __restrict__ P, const float* __restrict__ G22,
    float* __restrict__ eps, int Ldim)
{
  __shared__ float eps_s[8][1024];
  __shared__ float row_s[1024];
  const int tid = threadIdx.x;
  const int wave = tid >> 5, lane = tid & 31;
  const int b = blockIdx.x * 8 + wave;
  const float* pb = P + (size_t)b * Ldim;
  float* eb = eps + (size_t)b * Ldim;
  for (int i = 0; i < Ldim; ++i) {
    for (int j = tid; j < i; j += 256) row_s[j] = G22[(size_t)i * Ldim + j];
    __syncthreads();
    float part = 0.0f;
    for (int j = lane; j < i; j += 32) part += eps_s[wave][j] * row_s[j];
    #pragma unroll
    for (int off = 16; off > 0; off >>= 1) part += __shfl_xor(part, off, 32);
    float lam = 0.5f * (G22[(size_t)i * Ldim + i] + 1e-3f);
    float e = tanhf((pb[i] - part) / lam);
    if (lane == 0) { eps_s[wave][i] = e; eb[i] = e; }
    __syncthreads();
  }
}

// ---------------------------------------------------------------------------
// Blocked LU (no pivoting; E's symmetric part is PD)
// ---------------------------------------------------------------------------
__global__ __launch_bounds__(256) void lu_panel_kernel(float* __restrict__ E, int n, int k0)
{
  __shared__ float urow[64];
  const int tid = threadIdx.x;
  for (int j = 0; j < 64; ++j) {
    int jg = k0 + j;
    if (tid < 64) urow[tid] = E[(size_t)jg * n + k0 + tid];
    __syncthreads();
    float piv = urow[j];
    for (int i = jg + 1 + tid; i < n; i += 256) {
      float l = E[(size_t)i * n + jg] / piv;
      E[(size_t)i * n + jg] = l;
      for (int jj = j + 1; jj < 64; ++jj)
        E[(size_t)i * n + k0 + jj] -= l * urow[jj];
    }
    __syncthreads();
  }
}

__global__ __launch_bounds__(256) void lu_trsm_row_kernel(
    float* __restrict__ E, int n, int k0, int c0, int ncols)
{
  __shared__ float Lkk[64 * 64];
  const int tid = threadIdx.x;
  for (int idx = tid; idx < 64 * 64; idx += 256)
    Lkk[idx] = E[(size_t)(k0 + (idx >> 6)) * n + k0 + (idx & 63)];
  __syncthreads();
  int c = c0 + blockIdx.x * 256 + tid;
  if (c >= c0 + ncols) return;
  float a[64];
  #pragma unroll
  for (int r = 0; r < 64; ++r) a[r] = E[(size_t)(k0 + r) * n + c];
  #pragma unroll
  for (int r = 1; r < 64; ++r) {
    float s = a[r];
    #pragma unroll
    for (int q = 0; q < r; ++q) s -= Lkk[r * 64 + q] * a[q];
    a[r] = s;
  }
  #pragma unroll
  for (int r = 0; r < 64; ++r) E[(size_t)(k0 + r) * n + c] = a[r];
}

__global__ __launch_bounds__(256) void trsm_right_lowerT_unit_kernel(
    float* __restrict__ Xb, int ldx, const float* __restrict__ E, int n, int j0, int nrows)
{
  __shared__ float Ekk[64 * 64];
  const int tid = threadIdx.x;
  for (int idx = tid; idx < 64 * 64; idx += 256)
    Ekk[idx] = E[(size_t)(j0 + (idx >> 6)) * n + j0 + (idx & 63)];
  __syncthreads();
  int row = blockIdx.x * 256 + tid;
  if (row >= nrows) return;
  float* xr = Xb + (size_t)row * ldx + j0;
  float x[64];
  #pragma unroll
  for (int j = 0; j < 64; ++j) {
    float s = xr[j];
    #pragma unroll
    for (int q = 0; q < j; ++q) s -= x[q] * Ekk[j * 64 + q];
    x[j] = s;
  }
  #pragma unroll
  for (int j = 0; j < 64; ++j) xr[j] = x[j];
}

__global__ __launch_bounds__(256) void trsm_right_upperT_kernel(
    float* __restrict__ Xb, int ldx, const float* __restrict__ E, int n, int j0, int nrows)
{
  __shared__ float Ekk[64 * 64];
  const int tid = threadIdx.x;
  for (int idx = tid; idx < 64 * 64; idx += 256)
    Ekk[idx] = E[(size_t)(j0 + (idx >> 6)) * n + j0 + (idx & 63)];
  __syncthreads();
  int row = blockIdx.x * 256 + tid;
  if (row >= nrows) return;
  float* xr = Xb + (size_t)row * ldx + j0;
  float x[64];
  #pragma unroll
  for (int jj = 0; jj < 64; ++jj) {
    int j = 63 - jj;
    float s = xr[j];
    #pragma unroll
    for (int q = j + 1; q < 64; ++q) s -= x[q] * Ekk[j * 64 + q];
    x[j] = s / Ekk[j * 64 + j];
  }
  #pragma unroll
  for (int j = 0; j < 64; ++j) xr[j] = x[j];
}

// ---------------------------------------------------------------------------
extern "C" void kernel_launch(void* const* d_in, const int* in_sizes, int n_in,
                              void* d_out, int out_size, void* d_ws, size_t ws_size,
                              hipStream_t stream) {
  (void)in_sizes; (void)n_in; (void)out_size;
  const int B = 2048, NXI = 1024, LL = 1024, MM = 64;
  const float AMP = 20.0f;

  const float* y_   = (const float*)d_in[1];
  const float* xi   = (const float*)d_in[2];
  const float* u_om = (const float*)d_in[3];
  const float* x_ct = (const float*)d_in[4];
  const float* X    = (const float*)d_in[5];
  const float* Yp   = (const float*)d_in[6];
  const float* B2m  = (const float*)d_in[7];
  const float* C2m  = (const float*)d_in[8];
  const float* D21m = (const float*)d_in[9];
  const float* D22m = (const float*)d_in[10];
  const float* D12m = (const float*)d_in[11];
  const float* Am   = (const float*)d_in[12];
  const float* Bcm  = (const float*)d_in[13];
  const float* Cm   = (const float*)d_in[14];
  const float* Dm   = (const float*)d_in[15];

  float* out_u    = (float*)d_out;                    // (2048, 64)
  float* out_xi   = out_u + (size_t)B * MM;           // (2048, 1024)
  float* out_xctl = out_xi + (size_t)B * NXI;         // (2048, 64)

  if (ws_size < (size_t)9568256 * 4) return;          // 38.3 MB workspace
  float* ws   = (float*)d_ws;
  float* G21  = ws + 0 * 1048576;
  float* G22  = ws + 1 * 1048576;
  float* G31  = ws + 2 * 1048576;
  float* G32  = ws + 3 * 1048576;
  float* PX   = ws + 4 * 1048576;   // 2M: G11+G33, then P(preT), then E_xi/solve
  float* G11  = PX;
  float* G33  = PX + 1048576;
  float* Emat = ws + 6 * 1048576;
  float* EPS  = ws + 7 * 1048576;   // 2M
  float* W    = ws + 9 * 1048576;   // 128K

  auto gemm = [&](const float* A, const float* Bp, float* Cp,
                  int M, int N, int K, int lda, int ldb, int ldc,
                  float alpha, float beta, int tA, int tB) {
    dim3 g((N + TILE_N - 1) / TILE_N, (M + TILE_M - 1) / TILE_M);
    if (tA) {
      if (tB) gemm_wmma_f32<1,1><<<g, 256, 0, stream>>>(A, Bp, Cp, M, N, K, lda, ldb, ldc, alpha, beta);
      else    gemm_wmma_f32<1,0><<<g, 256, 0, stream>>>(A, Bp, Cp, M, N, K, lda, ldb, ldc, alpha, beta);
    } else {
      if (tB) gemm_wmma_f32<0,1><<<g, 256, 0, stream>>>(A, Bp, Cp, M, N, K, lda, ldb, ldc, alpha, beta);
      else    gemm_wmma_f32<0,0><<<g, 256, 0, stream>>>(A, Bp, Cp, M, N, K, lda, ldb, ldc, alpha, beta);
    }
  };

  // ---- Gram blocks G_ij = Xp_i^T @ Xp_j (K = 3072): async b128 both operands ----
  gemm(X + 1024, X + 0,    G21, 1024, 1024, 3072, 3072, 3072, 1024, 1.f, 0.f, 1, 0);
  gemm(X + 1024, X + 1024, G22, 1024, 1024, 3072, 3072, 3072, 1024, 1.f, 0.f, 1, 0);
  gemm(X + 2048, X + 0,    G31, 1024, 1024, 3072, 3072, 3072, 1024, 1.f, 0.f, 1, 0);
  gemm(X + 2048, X + 1024, G32, 1024, 1024, 3072, 3072, 3072, 1024, 1.f, 0.f, 1, 0);
  gemm(X + 0,    X + 0,    G11, 1024, 1024, 3072, 3072, 3072, 1024, 1.f, 0.f, 1, 0);
  gemm(X + 2048, X + 2048, G33, 1024, 1024, 3072, 3072, 3072, 1024, 1.f, 0.f, 1, 0);
  build_E_kernel<<<4096, 256, 0, stream>>>(Emat, G11, G33, Yp, 1024);

  // ---- w = y_ - x_ctl @ C^T - u_omega @ D^T ----
  gemm(x_ct, Cm, W, B, MM, MM, 64, 64, 64, -1.f, 0.f, 0, 1);
  gemm(u_om, Dm, W, B, MM, MM, 64, 64, 64, -1.f, 1.f, 0, 1);
  add_inplace_kernel<<<512, 256, 0, stream>>>(W, y_, B * MM);

  // ---- x_ctl_ = x_ctl @ A^T + u_omega @ Bc^T ----
  gemm(x_ct, Am,  out_xctl, B, MM, MM, 64, 64, 64, 1.f, 0.f, 0, 1);
  gemm(u_om, Bcm, out_xctl, B, MM, MM, 64, 64, 64, 1.f, 1.f, 0, 1);

  // ---- P = xi @ C1^T + w @ D12^T   (C1 = -G21) ----
  gemm(xi, G21,  PX, B, LL, NXI, 1024, 1024, 1024, -1.f, 0.f, 0, 1);
  gemm(W,  D12m, PX, B, LL, 64,   64,   64,   1024,  1.f, 1.f, 0, 1);

  // ---- tanh forward recurrence -> eps ----
  scan_eps_kernel<<<B / 8, 256, 0, stream>>>(PX, G22, EPS, LL);

  // ---- E_xi = xi @ Fm^T + eps @ B1^T + w @ B2^T  (Fm=G31, B1=G32) ----
  gemm(xi,  G31, PX, B, NXI, NXI, 1024, 1024, 1024, 1.f, 0.f, 0, 1);
  gemm(EPS, G32, PX, B, NXI, LL,  1024, 1024, 1024, 1.f, 1.f, 0, 1);
  gemm(W,   B2m, PX, B, NXI, 64,  64,   64,   1024, 1.f, 1.f, 0, 1);

  // ---- LU factor E (blocked NB=64; trailing updates on WMMA) ----
  for (int kb = 0; kb < 16; ++kb) {
    int k0 = kb * 64;
    lu_panel_kernel<<<1, 256, 0, stream>>>(Emat, 1024, k0);
    int rem = 1024 - k0 - 64;
    if (rem > 0) {
      lu_trsm_row_kernel<<<(rem + 255) / 256, 256, 0, stream>>>(Emat, 1024, k0, k0 + 64, rem);
      gemm(Emat + (size_t)(k0 + 64) * 1024 + k0,
           Emat + (size_t)k0 * 1024 + (k0 + 64),
           Emat + (size_t)(k0 + 64) * 1024 + (k0 + 64),
           rem, rem, 64, 1024, 1024, 1024, -1.f, 1.f, 0, 0);
    }
  }

  // ---- xi_ = E_xi @ E^{-T}: X L^T = E_xi (forward over col blocks) ----
  for (int jb = 0; jb < 16; ++jb) {
    int j0 = jb * 64;
    if (j0 > 0)
      gemm(PX, Emat + (size_t)j0 * 1024, PX + j0,
           B, 64, j0, 1024, 1024, 1024, -1.f, 1.f, 0, 1);
    trsm_right_lowerT_unit_kernel<<<8, 256, 0, stream>>>(PX, 1024, Emat, 1024, j0, B);
  }
  // ---- then X U^T = X (backward over col blocks) ----
  for (int jb = 15; jb >= 0; --jb) {
    int j0 = jb * 64;
    int k2 = 1024 - j0 - 64;
    if (k2 > 0)
      gemm(PX + j0 + 64, Emat + (size_t)j0 * 1024 + (j0 + 64), PX + j0,
           B, 64, k2, 1024, 1024, 1024, -1.f, 1.f, 0, 1);
    trsm_right_upperT_kernel<<<8, 256, 0, stream>>>(PX, 1024, Emat, 1024, j0, B);
  }
  copy_kernel<<<8192, 256, 0, stream>>>(out_xi, PX, B * NXI);

  // ---- u_ = AMP * (xi @ C2^T + eps @ D21^T + w @ D22^T) ----
  gemm(xi,  C2m,  out_u, B, MM, NXI, 1024, 1024, 64, AMP, 0.f, 0, 1);
  gemm(EPS, D21m, out_u, B, MM, LL,  1024, 1024, 64, AMP, 1.f, 0, 1);
  gemm(W,   D22m, out_u, B, MM, 64,  64,   64,   64, AMP, 1.f, 0, 1);
}